// SSIM_27436251086900
// MI455X (gfx1250) — compile-verified
//
#include <hip/hip_runtime.h>

// ---------------------------------------------------------------------------
// SSIM (11x11 Gaussian, sigma=1.5) on 16x3x512x512 fp32, fully fused:
// separable depthwise conv done as V_WMMA_F32_16X16X4_F32 tile matmuls,
// SSIM map + mean reduction in the same pass. Memory-bound workload
// (~100MB in / scalar out); fp32 kept end-to-end.
// ---------------------------------------------------------------------------

typedef __attribute__((ext_vector_type(2))) float v2f;
typedef __attribute__((ext_vector_type(8))) float v8f;

#define IMG_H 512
#define IMG_W 512
#define PLANES 48                       // 16 batch * 3 channels
#define TILES_X (IMG_W / 16)            // 32
#define TILES_Y (IMG_H / 16)            // 32
#define TILES_PER_PLANE (TILES_X * TILES_Y)      // 1024
#define TOTAL_TILES (PLANES * TILES_PER_PLANE)   // 49152
#define WAVES_PER_BLOCK 4
#define BLOCK_THREADS (WAVES_PER_BLOCK * 32)     // 128
#define NUM_BLOCKS (TOTAL_TILES / WAVES_PER_BLOCK) // 12288
#define HSTRIDE 20                      // LDS row stride (floats): bank-conflict free

// Normalized 11-tap Gaussian, sigma = 1.5 (precomputed on host side math).
__device__ __constant__ float GW[11] = {
    1.0283800e-03f, 7.5987582e-03f, 3.6000773e-02f, 1.0936063e-01f,
    2.1300553e-01f, 2.6601171e-01f, 2.1300553e-01f, 1.0936063e-01f,
    3.6000773e-02f, 7.5987582e-03f, 1.0283800e-03f};

__device__ __forceinline__ float wval(int i) {
  return (i >= 0 && i < 11) ? GW[i] : 0.0f;
}

__global__ __launch_bounds__(BLOCK_THREADS) void ssim_wmma_kernel(
    const float* __restrict__ img1, const float* __restrict__ img2,
    float* __restrict__ partial) {
  // Wave-private h buffers: [wave][channel][32 rows][stride 20]
  __shared__ float hbuf[WAVES_PER_BLOCK][5][32][HSTRIDE];
  __shared__ float wsum[WAVES_PER_BLOCK];

  const int lane = threadIdx.x & 31;
  const int wave = threadIdx.x >> 5;
  const int tile = blockIdx.x * WAVES_PER_BLOCK + wave;

  const int plane = tile / TILES_PER_PLANE;
  const int trem  = tile % TILES_PER_PLANE;
  const int ty = trem / TILES_X;
  const int tx = trem % TILES_X;
  const int row0 = ty * 16;
  const int col0 = tx * 16;

  const int mn  = lane & 15;              // row M (A-layout) == col N (B/C-layout)
  const int klo = (lane < 16) ? 0 : 2;    // first of the two K slots this lane holds

  // Banded Gaussian weight chunks.  Wh[c,x] = w(c-x) in B-layout and
  // Wv[m,j] = w(j-m) in A-layout reduce to the *same* per-lane values:
  //   chunk c, slot s:  wval(4c + klo + s - mn)
  v2f Wc[7];
#pragma unroll
  for (int c = 0; c < 7; ++c) {
    Wc[c].x = wval(4 * c + klo + 0 - mn);
    Wc[c].y = wval(4 * c + klo + 1 - mn);
  }

  const float* p1 = img1 + (size_t)plane * (IMG_H * IMG_W);
  const float* p2 = img2 + (size_t)plane * (IMG_H * IMG_W);

  // ---- Horizontal pass: two 16-row chunks covering h rows 0..31 (0..25 used)
#pragma unroll
  for (int rc = 0; rc < 2; ++rc) {
    const int r    = row0 + rc * 16 + mn - 5;       // image row for this lane's A rows
    const int rcl  = min(max(r, 0), IMG_H - 1);
    const float rm = (r >= 0 && r < IMG_H) ? 1.0f : 0.0f;

    v2f a1[7], a2[7];
#pragma unroll
    for (int c = 0; c < 7; ++c) {
      const int c0  = col0 + 4 * c + klo - 5;
      const int c1  = c0 + 1;
      const int c0c = min(max(c0, 0), IMG_W - 1);
      const int c1c = min(max(c1, 0), IMG_W - 1);
      const float m0 = (c0 >= 0 && c0 < IMG_W) ? rm : 0.0f;
      const float m1 = (c1 >= 0 && c1 < IMG_W) ? rm : 0.0f;
      a1[c].x = m0 * p1[rcl * IMG_W + c0c];
      a1[c].y = m1 * p1[rcl * IMG_W + c1c];
      a2[c].x = m0 * p2[rcl * IMG_W + c0c];
      a2[c].y = m1 * p2[rcl * IMG_W + c1c];
    }

#pragma unroll
    for (int ch = 0; ch < 5; ++ch) {
      v8f h = {};
#pragma unroll
      for (int c = 0; c < 7; ++c) {
        v2f A;
        if      (ch == 0) A = a1[c];
        else if (ch == 1) A = a2[c];
        else if (ch == 2) A = a1[c] * a1[c];
        else if (ch == 3) A = a2[c] * a2[c];
        else              A = a1[c] * a2[c];
        h = __builtin_amdgcn_wmma_f32_16x16x4_f32(
            false, A, false, Wc[c], (short)0, h, false, false);
      }
      // C-layout -> LDS image order: lane holds column mn, rows v / v+8.
#pragma unroll
      for (int v = 0; v < 8; ++v) {
        const int rr = rc * 16 + ((lane < 16) ? v : v + 8);
        hbuf[wave][ch][rr][mn] = h[v];
      }
    }
  }

  // ---- Vertical pass: out = Wv(16x28) x h(28x16) per channel
  v8f vacc[5];
#pragma unroll
  for (int ch = 0; ch < 5; ++ch) {
    v8f acc = {};
#pragma unroll
    for (int c = 0; c < 7; ++c) {
      v2f B;  // B-layout: lane = column mn, VGPR slots = K rows 4c+klo, 4c+klo+1
      B.x = hbuf[wave][ch][4 * c + klo + 0][mn];
      B.y = hbuf[wave][ch][4 * c + klo + 1][mn];
      acc = __builtin_amdgcn_wmma_f32_16x16x4_f32(
          false, Wc[c], false, B, (short)0, acc, false, false);
    }
    vacc[ch] = acc;
  }

  // ---- SSIM map on the five identically-laid-out accumulators, then reduce
  const float C1 = 0.0001f;   // 0.01^2
  const float C2 = 0.0009f;   // 0.03^2
  float s = 0.0f;
#pragma unroll
  for (int v = 0; v < 8; ++v) {
    const float mu1 = vacc[0][v], mu2 = vacc[1][v];
    const float x11 = vacc[2][v], x22 = vacc[3][v], x12 = vacc[4][v];
    const float mu1s = mu1 * mu1, mu2s = mu2 * mu2, mu12 = mu1 * mu2;
    const float s1 = x11 - mu1s, s2 = x22 - mu2s, s12 = x12 - mu12;
    const float num = (2.0f * mu12 + C1) * (2.0f * s12 + C2);
    const float den = (mu1s + mu2s + C1) * (s1 + s2 + C2);
    s += num / den;
  }
#pragma unroll
  for (int off = 16; off > 0; off >>= 1) s += __shfl_xor(s, off, 32);
  if (lane == 0) wsum[wave] = s;
  __syncthreads();
  if (threadIdx.x == 0) {
    float t = 0.0f;
#pragma unroll
    for (int w = 0; w < WAVES_PER_BLOCK; ++w) t += wsum[w];
    partial[blockIdx.x] = t;
  }
}

__global__ __launch_bounds__(256) void ssim_reduce_kernel(
    const float* __restrict__ partial, float* __restrict__ out) {
  __shared__ float sm[256];
  float s = 0.0f;
  for (int i = threadIdx.x; i < NUM_BLOCKS; i += 256) s += partial[i];
  sm[threadIdx.x] = s;
  __syncthreads();
  for (int off = 128; off > 0; off >>= 1) {
    if ((int)threadIdx.x < off) sm[threadIdx.x] += sm[threadIdx.x + off];
    __syncthreads();
  }
  if (threadIdx.x == 0)
    out[0] = sm[0] * (1.0f / (float)(PLANES * IMG_H * IMG_W));
}

extern "C" void kernel_launch(void* const* d_in, const int* in_sizes, int n_in,
                              void* d_out, int out_size, void* d_ws,
                              size_t ws_size, hipStream_t stream) {
  const float* img1 = (const float*)d_in[0];
  const float* img2 = (const float*)d_in[1];
  float* partial = (float*)d_ws;  // NUM_BLOCKS floats (48 KB)

  ssim_wmma_kernel<<<NUM_BLOCKS, BLOCK_THREADS, 0, stream>>>(img1, img2,
                                                             partial);
  ssim_reduce_kernel<<<1, 256, 0, stream>>>(partial, (float*)d_out);
}